// SHGNN_88175678587255
// MI455X (gfx1250) — compile-verified
//
#include <hip/hip_runtime.h>
#include <hip/hip_bf16.h>

#define N_NODES  100000
#define N_HEDGES 50000
#define ME       400000
#define MN       400000
#define D_IN     128
#define D_HID    64

typedef __attribute__((ext_vector_type(2))) float v2f;
typedef __attribute__((ext_vector_type(8))) float v8f;

// ---------------------------------------------------------------- helpers
__device__ __forceinline__ unsigned ord_f32(float x) {
    unsigned b = __float_as_uint(x);
    return (b & 0x80000000u) ? ~b : (b | 0x80000000u);
}
__device__ __forceinline__ float unord_f32(unsigned u) {
    unsigned b = (u & 0x80000000u) ? (u ^ 0x80000000u) : ~u;
    return __uint_as_float(b);
}

// ---------------------------------------------------------------- K/V GEMM
// K = X@Wk + bk, V = X@Wv + bv.  X = [xA (dA cols) | xB (dB cols)] per row.
// Block = 128 threads (4 waves); block computes a 16 x 64 tile of K and V.
// A-tile staged in LDS via coalesced loads; WMMA = V_WMMA_F32_16X16X4_F32.
// Launch invariant: M % 16 == 0 (100000 / 50000 both are).
__global__ void shgnn_gemm_kv(const float* __restrict__ xA, int dA,
                              const float* __restrict__ xB, int dB,
                              const float* __restrict__ Wk, const float* __restrict__ bk,
                              const float* __restrict__ Wv, const float* __restrict__ bv,
                              float* __restrict__ K, float* __restrict__ V) {
    __shared__ __align__(16) float sa[16][192];     // A tile, din <= 192
    const int din  = dA + dB;
    const int tid  = threadIdx.x;
    const int lane = tid & 31;
    const int wave = tid >> 5;
    const int hi   = lane >> 4;      // 0: K {0,1} , 1: K {2,3}
    const int l15  = lane & 15;
    const int rowBase = blockIdx.x * 16;
    const int col = wave * 16 + l15;

    // coalesced cooperative stage of the 16 x din A tile
    for (int r = 0; r < 16; ++r) {
        const float* ra = xA + (size_t)(rowBase + r) * dA;
        for (int c = tid; c < dA; c += 128) sa[r][c] = ra[c];
        if (dB > 0) {
            const float* rbp = xB + (size_t)(rowBase + r) * dB;
            for (int c = tid; c < dB; c += 128) sa[r][dA + c] = rbp[c];
        }
    }
    __syncthreads();

    v8f ck = {}; v8f cv = {};
    for (int kk = 0; kk < din; kk += 4) {
        const int k0 = kk + hi * 2;                 // k0 even -> 8B aligned
        const v2f a = *(const v2f*)&sa[l15][k0];
        v2f bK, bV;
        bK.x = Wk[(k0    ) * 64 + col]; bK.y = Wk[(k0 + 1) * 64 + col];
        bV.x = Wv[(k0    ) * 64 + col]; bV.y = Wv[(k0 + 1) * 64 + col];
        ck = __builtin_amdgcn_wmma_f32_16x16x4_f32(false, a, false, bK, (short)0, ck, false, false);
        cv = __builtin_amdgcn_wmma_f32_16x16x4_f32(false, a, false, bV, (short)0, cv, false, false);
    }
    const float bkc = bk[col], bvc = bv[col];
    for (int i = 0; i < 8; ++i) {
        const int r = rowBase + i + hi * 8;
        K[(size_t)r * 64 + col] = ck[i] + bkc;
        V[(size_t)r * 64 + col] = cv[i] + bvc;
    }
}

// ---------------------------------------------------------------- alpha = K . seed
__global__ void shgnn_alpha(const float* __restrict__ K, const float* __restrict__ seed,
                            float* __restrict__ alpha, int M) {
    const int i = blockIdx.x * blockDim.x + threadIdx.x;
    if (i >= M) return;
    const float4* kr = (const float4*)(K + (size_t)i * 64);
    const float4* sd = (const float4*)seed;
    float acc = 0.f;
    for (int c = 0; c < 16; ++c) {
        const float4 k = kr[c], s = sd[c];
        acc += k.x * s.x + k.y * s.y + k.z * s.z + k.w * s.w;
    }
    alpha[i] = acc;
}

// ---------------------------------------------------------------- segment softmax (3 passes)
__global__ void shgnn_seg_max(const int* __restrict__ srcA, const int* __restrict__ batch,
                              const int* __restrict__ map, const float* __restrict__ alpha,
                              unsigned* __restrict__ segmax, int M) {
    const int j = blockIdx.x * blockDim.x + threadIdx.x;
    if (j >= 2 * M) return;
    const int s   = (j < M) ? srcA[j] : (j - M);
    const int seg = batch[s];
    const float a = alpha[map[s]];
    atomicMax(&segmax[seg], ord_f32(a));
}

__global__ void shgnn_seg_sum(const int* __restrict__ srcA, const int* __restrict__ batch,
                              const int* __restrict__ map, const float* __restrict__ alpha,
                              const unsigned* __restrict__ segmax, float* __restrict__ segsum,
                              const float* __restrict__ V, int M) {
    const int j = blockIdx.x * blockDim.x + threadIdx.x;
    if (j >= 2 * M) return;
    const int s    = (j < M) ? srcA[j] : (j - M);
    const int seg  = batch[s];
    const int vrow = map[s];
    const float a  = alpha[vrow];
    const float m  = unord_f32(segmax[seg]);
    atomicAdd(&segsum[seg], __expf(a - m));
    // warm L2 for the weighted-V scatter pass that follows (global_prefetch_b8)
    __builtin_prefetch(V + (size_t)vrow * 64, 0, 1);
    __builtin_prefetch(V + (size_t)vrow * 64 + 32, 0, 1);
}

// One wave per (src,seg) pair; lanes cover the 64 V columns coalesced.
__global__ void shgnn_scatter_v(const int* __restrict__ srcA, const int* __restrict__ batch,
                                const int* __restrict__ map, const float* __restrict__ alpha,
                                const unsigned* __restrict__ segmax, const float* __restrict__ segsum,
                                const float* __restrict__ V, float* __restrict__ segraw, int M) {
    const int lane = threadIdx.x & 31;
    const int j = blockIdx.x * (blockDim.x >> 5) + (threadIdx.x >> 5);
    if (j >= 2 * M) return;
    const int s    = (j < M) ? srcA[j] : (j - M);
    const int seg  = batch[s];
    const int vrow = map[s];
    const float a  = alpha[vrow];
    const float m  = unord_f32(segmax[seg]);
    const float w  = __expf(a - m) / segsum[seg];
    const float* vr = V + (size_t)vrow * 64;
    float* o = segraw + (size_t)seg * 64;
    atomicAdd(&o[lane],      w * vr[lane]);
    atomicAdd(&o[lane + 32], w * vr[lane + 32]);
}

// ---------------------------------------------------------------- PMA tail:
// out = LN(raw+seed); ff = relu(out@W1+b1)@W2+b2; res = relu(LN(out+relu(ff)))
// Block = 128 threads handles 16 segment rows fully fused in LDS; 2x 64x64 WMMA GEMM.
// Launch invariant: S % 16 == 0.
__global__ void shgnn_pma_post(const float* __restrict__ raw,
                               const float* __restrict__ seed,
                               const float* __restrict__ g0, const float* __restrict__ c0,
                               const float* __restrict__ W1, const float* __restrict__ b1,
                               const float* __restrict__ W2, const float* __restrict__ b2,
                               const float* __restrict__ g1, const float* __restrict__ c1,
                               float* __restrict__ outp) {
    __shared__ __align__(16) float lx[16][64];
    __shared__ __align__(16) float lo[16][64];
    __shared__ __align__(16) float lh[16][64];
    __shared__ float mu[16], rs[16];
    const int tid  = threadIdx.x;
    const int lane = tid & 31, wave = tid >> 5;
    const int hi = lane >> 4, l15 = lane & 15;
    const int rb = blockIdx.x * 16;

    for (int e = tid; e < 16 * 64; e += 128) {
        const int r = e >> 6, c = e & 63;
        lx[r][c] = raw[(size_t)(rb + r) * 64 + c] + seed[c];
    }
    __syncthreads();
    if (tid < 16) {
        float m = 0.f; for (int c = 0; c < 64; ++c) m += lx[tid][c]; m *= (1.f / 64.f);
        float v = 0.f; for (int c = 0; c < 64; ++c) { float d = lx[tid][c] - m; v += d * d; }
        mu[tid] = m; rs[tid] = rsqrtf(v * (1.f / 64.f) + 1e-5f);
    }
    __syncthreads();
    for (int e = tid; e < 16 * 64; e += 128) {
        const int r = e >> 6, c = e & 63;
        lo[r][c] = (lx[r][c] - mu[r]) * rs[r] * g0[c] + c0[c];
    }
    __syncthreads();

    const int col = wave * 16 + l15;
    v8f acc = {};
    for (int kk = 0; kk < 64; kk += 4) {
        const int k0 = kk + hi * 2;
        const v2f a = *(const v2f*)&lo[l15][k0];
        v2f b;
        b.x = W1[k0 * 64 + col]; b.y = W1[(k0 + 1) * 64 + col];
        acc = __builtin_amdgcn_wmma_f32_16x16x4_f32(false, a, false, b, (short)0, acc, false, false);
    }
    {
        const float bb = b1[col];
        for (int i = 0; i < 8; ++i) lh[i + hi * 8][col] = fmaxf(acc[i] + bb, 0.f);
    }
    __syncthreads();

    acc = (v8f){};
    for (int kk = 0; kk < 64; kk += 4) {
        const int k0 = kk + hi * 2;
        const v2f a = *(const v2f*)&lh[l15][k0];
        v2f b;
        b.x = W2[k0 * 64 + col]; b.y = W2[(k0 + 1) * 64 + col];
        acc = __builtin_amdgcn_wmma_f32_16x16x4_f32(false, a, false, b, (short)0, acc, false, false);
    }
    {
        const float bb = b2[col];
        for (int i = 0; i < 8; ++i) {
            const int r = i + hi * 8;
            lx[r][col] = lo[r][col] + fmaxf(acc[i] + bb, 0.f);
        }
    }
    __syncthreads();
    if (tid < 16) {
        float m = 0.f; for (int c = 0; c < 64; ++c) m += lx[tid][c]; m *= (1.f / 64.f);
        float v = 0.f; for (int c = 0; c < 64; ++c) { float d = lx[tid][c] - m; v += d * d; }
        mu[tid] = m; rs[tid] = rsqrtf(v * (1.f / 64.f) + 1e-5f);
    }
    __syncthreads();
    for (int e = tid; e < 16 * 64; e += 128) {
        const int r = e >> 6, c = e & 63;
        const float v = (lx[r][c] - mu[r]) * rs[r] * g1[c] + c1[c];
        outp[(size_t)(rb + r) * 64 + c] = fmaxf(v, 0.f);   // outer relu
    }
}

// ---------------------------------------------------------------- output head:
// x=[node_x|nx0|nx1] (256) -> relu(x@W1+b1) (512) -> @W2+b2 (40) -> log_softmax
// Launch invariant: M % 16 == 0.
__global__ void shgnn_out_head(const float* __restrict__ node_x,
                               const float* __restrict__ nx0, const float* __restrict__ nx1,
                               const float* __restrict__ W1, const float* __restrict__ b1,
                               const float* __restrict__ W2, const float* __restrict__ b2,
                               float* __restrict__ out, int M) {
    __shared__ __align__(16) float lx[16][256];
    __shared__ __align__(16) float lh[4][16][18];   // pad 18: keeps v2f reads 8B aligned
    __shared__ __align__(16) float lg[4][16][48];
    const int tid = threadIdx.x;
    const int lane = tid & 31, wave = tid >> 5;
    const int hi = lane >> 4, l15 = lane & 15;
    const int rb = blockIdx.x * 16;

    for (int e = tid; e < 16 * 256; e += 128) {
        const int r = e >> 8, c = e & 255;
        const int gr = rb + r;
        float v;
        if (c < 128)      v = node_x[(size_t)gr * 128 + c];
        else if (c < 192) v = nx0[(size_t)gr * 64 + (c - 128)];
        else              v = nx1[(size_t)gr * 64 + (c - 192)];
        lx[r][c] = v;
    }
    __syncthreads();

    v8f lgacc[3] = {};
    for (int ch = wave; ch < 32; ch += 4) {            // hidden chunks of 16
        v8f hacc = {};
        const int col1 = ch * 16 + l15;
        for (int kk = 0; kk < 256; kk += 4) {
            const int k0 = kk + hi * 2;
            const v2f a = *(const v2f*)&lx[l15][k0];
            v2f b;
            b.x = W1[(size_t)k0 * 512 + col1]; b.y = W1[(size_t)(k0 + 1) * 512 + col1];
            hacc = __builtin_amdgcn_wmma_f32_16x16x4_f32(false, a, false, b, (short)0, hacc, false, false);
        }
        const float bb = b1[col1];
        for (int i = 0; i < 8; ++i) lh[wave][i + hi * 8][l15] = fmaxf(hacc[i] + bb, 0.f);
        // wave-private LDS tile; DS ops are in-order within a wave -> no barrier needed
        for (int t = 0; t < 3; ++t) {
            const int col2 = t * 16 + l15;
            const int c2   = (col2 < 40) ? col2 : 39;      // clamp: unconditional loads
            const float msk = (col2 < 40) ? 1.f : 0.f;
            v8f a2 = lgacc[t];
            for (int kk = 0; kk < 16; kk += 4) {
                const int k0 = kk + hi * 2;
                const v2f a = *(const v2f*)&lh[wave][l15][k0];
                const int kr = ch * 16 + k0;
                v2f b;
                b.x = W2[(size_t)kr * 40 + c2] * msk;
                b.y = W2[(size_t)(kr + 1) * 40 + c2] * msk;
                a2 = __builtin_amdgcn_wmma_f32_16x16x4_f32(false, a, false, b, (short)0, a2, false, false);
            }
            lgacc[t] = a2;
        }
    }
    for (int t = 0; t < 3; ++t)
        for (int i = 0; i < 8; ++i)
            lg[wave][i + hi * 8][t * 16 + l15] = lgacc[t][i];
    __syncthreads();
    for (int e = tid; e < 16 * 48; e += 128) {
        const int r = e / 48, c = e % 48;
        lg[0][r][c] += lg[1][r][c] + lg[2][r][c] + lg[3][r][c];
    }
    __syncthreads();
    if (tid < 16) {
        const int gr = rb + tid;
        float mx = -3.4e38f;
        for (int c = 0; c < 40; ++c) {
            const float s = lg[0][tid][c] + b2[c];
            lg[0][tid][c] = s;
            mx = fmaxf(mx, s);
        }
        float se = 0.f;
        for (int c = 0; c < 40; ++c) se += __expf(lg[0][tid][c] - mx);
        const float ls = __logf(se);
        for (int c = 0; c < 40; ++c) out[(size_t)gr * 40 + c] = lg[0][tid][c] - mx - ls;
    }
}

// ---------------------------------------------------------------- launch
extern "C" void kernel_launch(void* const* d_in, const int* in_sizes, int n_in,
                              void* d_out, int out_size, void* d_ws, size_t ws_size,
                              hipStream_t stream) {
    (void)in_sizes; (void)n_in; (void)out_size; (void)ws_size;
    // Assumed flattening: top-level dict insertion order; params flattened
    // JAX-style (keys sorted: 'e2n','n2e','out'; per-PMA keys sorted:
    // W1,W2,Wk,Wv,b1,b2,bk,bv,c0,c1,g0,g1,seed).
    auto F = [&](int i) { return (const float*)d_in[i]; };
    struct Pma { const float *W1,*W2,*Wk,*Wv,*b1,*b2,*bk,*bv,*c0,*c1,*g0,*g1,*seed; };
    auto pma_at = [&](int b) {
        Pma p; p.W1=F(b); p.W2=F(b+1); p.Wk=F(b+2); p.Wv=F(b+3); p.b1=F(b+4); p.b2=F(b+5);
        p.bk=F(b+6); p.bv=F(b+7); p.c0=F(b+8); p.c1=F(b+9); p.g0=F(b+10); p.g1=F(b+11);
        p.seed=F(b+12); return p;
    };
    const float* node_x = F(0);
    Pma e2n[2] = { pma_at(1),  pma_at(14) };
    Pma n2e[2] = { pma_at(27), pma_at(40) };
    const float *oW1 = F(53), *oW2 = F(54), *ob1 = F(55), *ob2 = F(56);
    const int* nodes_map = (const int*)d_in[57];
    const int* eb_src    = (const int*)d_in[58];
    const int* eb_batch  = (const int*)d_in[59];
    const int* edges_map = (const int*)d_in[60];
    const int* nb_src    = (const int*)d_in[61];
    const int* nb_batch  = (const int*)d_in[62];

    // workspace carve-out (256B aligned blocks)
    float* ws = (float*)d_ws;
    size_t off = 0;
    auto take = [&](size_t n) { float* p = ws + off; off += (n + 63) & ~(size_t)63; return p; };
    float*    nx0      = take((size_t)N_NODES * 64);
    float*    nx1      = take((size_t)N_NODES * 64);
    float*    Kbuf     = take((size_t)N_NODES * 64);
    float*    Vbuf     = take((size_t)N_NODES * 64);
    float*    alphaBuf = take(N_NODES);
    unsigned* segmaxU  = (unsigned*)take(N_NODES);
    float*    segsum   = take(N_NODES);
    float*    segraw   = take((size_t)N_NODES * 64);
    float*    edgeX    = take((size_t)N_HEDGES * 64);

    float* nxbuf[2] = { nx0, nx1 };
    const int P = 2 * ME;   // == 2*MN

    for (int L = 0; L < 2; ++L) {
        // ---- N2E: K/V on unique node rows (gather commutes with GEMM)
        shgnn_gemm_kv<<<N_NODES / 16, 128, 0, stream>>>(
            node_x, 128, (L == 0) ? nullptr : nx0, (L == 0) ? 0 : 64,
            n2e[L].Wk, n2e[L].bk, n2e[L].Wv, n2e[L].bv, Kbuf, Vbuf);
        shgnn_alpha<<<(N_NODES + 255) / 256, 256, 0, stream>>>(Kbuf, n2e[L].seed, alphaBuf, N_NODES);
        hipMemsetAsync(segmaxU, 0, (size_t)N_HEDGES * 4, stream);
        hipMemsetAsync(segsum,  0, (size_t)N_HEDGES * 4, stream);
        hipMemsetAsync(segraw,  0, (size_t)N_HEDGES * 64 * 4, stream);
        shgnn_seg_max<<<(P + 255) / 256, 256, 0, stream>>>(eb_src, eb_batch, nodes_map, alphaBuf, segmaxU, ME);
        shgnn_seg_sum<<<(P + 255) / 256, 256, 0, stream>>>(eb_src, eb_batch, nodes_map, alphaBuf, segmaxU, segsum, Vbuf, ME);
        shgnn_scatter_v<<<(P + 7) / 8, 256, 0, stream>>>(eb_src, eb_batch, nodes_map, alphaBuf, segmaxU, segsum, Vbuf, segraw, ME);
        shgnn_pma_post<<<N_HEDGES / 16, 128, 0, stream>>>(
            segraw, n2e[L].seed, n2e[L].g0, n2e[L].c0, n2e[L].W1, n2e[L].b1,
            n2e[L].W2, n2e[L].b2, n2e[L].g1, n2e[L].c1, edgeX);

        // ---- E2N: K/V on unique hyperedge rows
        shgnn_gemm_kv<<<N_HEDGES / 16, 128, 0, stream>>>(
            edgeX, 64, nullptr, 0,
            e2n[L].Wk, e2n[L].bk, e2n[L].Wv, e2n[L].bv, Kbuf, Vbuf);
        shgnn_alpha<<<(N_HEDGES + 255) / 256, 256, 0, stream>>>(Kbuf, e2n[L].seed, alphaBuf, N_HEDGES);
        hipMemsetAsync(segmaxU, 0, (size_t)N_NODES * 4, stream);
        hipMemsetAsync(segsum,  0, (size_t)N_NODES * 4, stream);
        hipMemsetAsync(segraw,  0, (size_t)N_NODES * 64 * 4, stream);
        shgnn_seg_max<<<(P + 255) / 256, 256, 0, stream>>>(nb_src, nb_batch, edges_map, alphaBuf, segmaxU, MN);
        shgnn_seg_sum<<<(P + 255) / 256, 256, 0, stream>>>(nb_src, nb_batch, edges_map, alphaBuf, segmaxU, segsum, Vbuf, MN);
        shgnn_scatter_v<<<(P + 7) / 8, 256, 0, stream>>>(nb_src, nb_batch, edges_map, alphaBuf, segmaxU, segsum, Vbuf, segraw, MN);
        shgnn_pma_post<<<N_NODES / 16, 128, 0, stream>>>(
            segraw, e2n[L].seed, e2n[L].g0, e2n[L].c0, e2n[L].W1, e2n[L].b1,
            e2n[L].W2, e2n[L].b2, e2n[L].g1, e2n[L].c1, nxbuf[L]);
    }

    shgnn_out_head<<<N_NODES / 16, 128, 0, stream>>>(
        node_x, nx0, nx1, oW1, ob1, oW2, ob2, (float*)d_out, N_NODES);
}